// KeyValueBottleneck_63797444215631
// MI455X (gfx1250) — compile-verified
//
#include <hip/hip_runtime.h>
#include <hip/hip_bf16.h>
#include <stddef.h>

// ---------------------------------------------------------------------------
// KeyValueBottleneck for MI455X (gfx1250, wave32, WMMA + async global->LDS)
//   x:            (b=32, C=64, n=64, dk=16)  f32
//   keys:         (C, J=1024, dk)            f32
//   values:       (C, J, dv=16)              f32
//   keys_avg:     (C, J, dk)                 f32
//   cluster_size: (C, J)                     f32
// Outputs (flat, concatenated):
//   qv (b,C,1,n,dv) | new_keys (C,J,dk) | new_cluster_size (C,J) | new_keys_avg (C,J,dk)
// ---------------------------------------------------------------------------

typedef float v2f __attribute__((ext_vector_type(2)));
typedef float v8f __attribute__((ext_vector_type(8)));
typedef int   v4i __attribute__((ext_vector_type(4)));

// Pointer types matching the async-DMA builtin's signature
// (param0: AS1 non-const v4i*, param1: AS3 v4i*).
typedef v4i __attribute__((address_space(1))) kvb_g4;
typedef v4i __attribute__((address_space(3))) kvb_l4;

#define KVB_B  32
#define KVB_C  64
#define KVB_N  64
#define KVB_DK 16
#define KVB_DV 16
#define KVB_J  1024
#define KVB_I  (KVB_B * KVB_N)   // 2048 rows per channel
#define KVB_KCH 512              // key rows staged in LDS per chunk
#define KVB_KSTRIDE 20           // LDS row stride in floats (bank-conflict-free)
#define KVB_DECAY 0.95f
#define KVB_EPS   1e-5f

#if defined(__has_builtin)
#if __has_builtin(__builtin_amdgcn_global_load_async_to_lds_b128) && \
    __has_builtin(__builtin_amdgcn_s_wait_asynccnt)
#define KVB_ASYNC 1
#endif
#endif

// ---------------------------------------------------------------------------
// Kernel 1: per-channel nearest-key argmax via WMMA + value gather.
// Grid: (C, I/128). Block: 256 (8 waves). Keys staged chunk-wise into LDS via
// async global->LDS DMA; |k|^2 precomputed per row. Each wave owns a 16-row
// tile and scans keys in 16-column WMMA tiles with a fused running argmax.
// score(i,j) = 2*<f_i,k_j> - |k_j|^2   (the -|f_i|^2 term is argmax-invariant)
// ---------------------------------------------------------------------------
__global__ __launch_bounds__(256) void kvb_assign(
    const float* __restrict__ x,
    const float* __restrict__ keys,
    const float* __restrict__ values,
    float* __restrict__ qv_out,        // (b,C,1,n,dv)
    int* __restrict__ ind_ws)          // (C, I)
{
    __shared__ float skeys[KVB_KCH * KVB_KSTRIDE];   // 40 KB
    __shared__ float sknorm[KVB_KCH];                //  2 KB

    const int c    = blockIdx.x;                 // channel
    const int tile = blockIdx.y;                 // 16 row-tiles of 128
    const int t    = threadIdx.x;
    const int wave = t >> 5;                     // 0..7
    const int lane = t & 31;
    const int half = lane >> 4;                  // 0 | 1
    const int l15  = lane & 15;

    const int row   = tile * 128 + wave * 16 + l15;   // i in [0,2048)
    const int b_idx = row >> 6;                       // i / n
    const int n_idx = row & 63;                       // i % n
    const float* __restrict__ xrow =
        x + ((((size_t)b_idx * KVB_C + c) * KVB_N + n_idx) * KVB_DK);
    const float* __restrict__ kc = keys + (size_t)c * KVB_J * KVB_DK;

    // A fragments for the 4 chained K=4 WMMA steps (f32 16x16x4 layout):
    // lanes 0-15 hold k = 4q+{0,1} of row l15; lanes 16-31 hold k = 4q+{2,3}.
    v2f afr[4];
#pragma unroll
    for (int q = 0; q < 4; ++q)
        afr[q] = *(const v2f*)(xrow + 4 * q + 2 * half);

    float best[8];
    int   bidx[8];
#pragma unroll
    for (int m = 0; m < 8; ++m) { best[m] = -3.4e38f; bidx[m] = 0; }

    for (int ch = 0; ch < KVB_J / KVB_KCH; ++ch) {
        const float* __restrict__ kchunk = kc + (size_t)ch * KVB_KCH * KVB_DK;

        // ---- stage 512 key rows (64B each) into LDS, padded stride 80B ----
#if KVB_ASYNC
#pragma unroll
        for (int it = 0; it < (KVB_KCH * 4) / 256; ++it) {   // 8 x b128 / thread
            const int q = t + it * 256;                      // quarter-row id
            const int r = q >> 2, p = q & 3;
            __builtin_amdgcn_global_load_async_to_lds_b128(
                (kvb_g4*)(kchunk + r * KVB_DK + p * 4),
                (kvb_l4*)(skeys + r * KVB_KSTRIDE + p * 4),
                0, 0);
        }
#else
#pragma unroll
        for (int it = 0; it < (KVB_KCH * 4) / 256; ++it) {
            const int q = t + it * 256;
            const int r = q >> 2, p = q & 3;
            *(float4*)(skeys + r * KVB_KSTRIDE + p * 4) =
                *(const float4*)(kchunk + r * KVB_DK + p * 4);
        }
#endif
        // ---- |k|^2 per row (overlaps with the async DMA) ----
#pragma unroll
        for (int it = 0; it < KVB_KCH / 256; ++it) {         // 2 rows / thread
            const int r = t + it * 256;
            const float4* __restrict__ kr = (const float4*)(kchunk + r * KVB_DK);
            const float4 a0 = kr[0], a1 = kr[1], a2 = kr[2], a3 = kr[3];
            sknorm[r] =
                (a0.x*a0.x + a0.y*a0.y + a0.z*a0.z + a0.w*a0.w) +
                (a1.x*a1.x + a1.y*a1.y + a1.z*a1.z + a1.w*a1.w) +
                (a2.x*a2.x + a2.y*a2.y + a2.z*a2.z + a2.w*a2.w) +
                (a3.x*a3.x + a3.y*a3.y + a3.z*a3.z + a3.w*a3.w);
        }
#if KVB_ASYNC
        __builtin_amdgcn_s_wait_asynccnt(0);
#endif
        __syncthreads();

        // ---- WMMA scan over this chunk: 32 tiles of 16 keys ----
        for (int j0 = 0; j0 < KVB_KCH; j0 += 16) {
            const int r = j0 + l15;
            const float* __restrict__ srow = skeys + r * KVB_KSTRIDE + 2 * half;
            v2f bfr[4];
#pragma unroll
            for (int q = 0; q < 4; ++q)
                bfr[q] = *(const v2f*)(srow + 4 * q);        // ds_load_b64
            const float kn = sknorm[r];                      // ds_load_b32

            v8f acc = {0.f, 0.f, 0.f, 0.f, 0.f, 0.f, 0.f, 0.f};
#pragma unroll
            for (int q = 0; q < 4; ++q) {
                acc = __builtin_amdgcn_wmma_f32_16x16x4_f32(
                    false, afr[q], false, bfr[q], (short)0, acc, false, false);
            }

            // C/D layout: vgpr m <-> row M = half*8 + m; column N = l15.
            const int jj = ch * KVB_KCH + j0 + l15;
#pragma unroll
            for (int m = 0; m < 8; ++m) {
                const float s = 2.f * acc[m] - kn;
                if (s > best[m]) { best[m] = s; bidx[m] = jj; } // strict > => first max
            }
        }
        __syncthreads();   // all waves done with this chunk before re-staging
    }

    // Reduce argmax across the 16 lanes of each half (xor stays within half).
#pragma unroll
    for (int m = 0; m < 8; ++m) {
#pragma unroll
        for (int mask = 1; mask < 16; mask <<= 1) {
            const float ov = __shfl_xor(best[m], mask, 32);
            const int   oi = __shfl_xor(bidx[m], mask, 32);
            if (ov > best[m] || (ov == best[m] && oi < bidx[m])) {
                best[m] = ov;
                bidx[m] = oi;
            }
        }
    }

    // Lanes {0..7} of each half emit rows M = half*8 + m.
    if (l15 < 8) {
        const int m     = l15;
        const int rrow  = tile * 128 + wave * 16 + half * 8 + m;
        const int jstar = bidx[m];
        ind_ws[c * KVB_I + rrow] = jstar;

        const int rb = rrow >> 6;
        const int rn = rrow & 63;
        const float4* __restrict__ vsrc =
            (const float4*)(values + ((size_t)c * KVB_J + jstar) * KVB_DV);
        float4* __restrict__ vdst =
            (float4*)(qv_out + (((size_t)rb * KVB_C + c) * KVB_N + rn) * KVB_DV);
        vdst[0] = vsrc[0];
        vdst[1] = vsrc[1];
        vdst[2] = vsrc[2];
        vdst[3] = vsrc[3];
    }
}

// ---------------------------------------------------------------------------
// Kernel 2: deterministic segment count/sum + EMA update (no float atomics).
// Grid: (C, J/256). Block: 256. Assignment indices staged in LDS (broadcast
// reads); each thread owns one codebook entry j and scans all 2048 rows.
// ---------------------------------------------------------------------------
__global__ __launch_bounds__(256) void kvb_segsum(
    const float* __restrict__ x,
    const float* __restrict__ keys_avg,
    const float* __restrict__ cluster_size,
    const int* __restrict__ ind_ws,
    float* __restrict__ out_cs,        // (C, J)
    float* __restrict__ out_ka)        // (C, J, dk)
{
    __shared__ int sind[KVB_I];
    const int c = blockIdx.x;
    const int j = blockIdx.y * 256 + threadIdx.x;

    for (int i = threadIdx.x; i < KVB_I; i += 256)
        sind[i] = ind_ws[c * KVB_I + i];
    __syncthreads();

    float cnt = 0.f;
    float s[KVB_DK];
#pragma unroll
    for (int k = 0; k < KVB_DK; ++k) s[k] = 0.f;

    for (int i = 0; i < KVB_I; ++i) {
        if (sind[i] == j) {                       // rare: ~2 hits per thread
            cnt += 1.f;
            const float* __restrict__ xr =
                x + (((size_t)(i >> 6) * KVB_C + c) * KVB_N + (i & 63)) * KVB_DK;
#pragma unroll
            for (int k = 0; k < KVB_DK; ++k) s[k] += xr[k];
        }
    }

    out_cs[(size_t)c * KVB_J + j] =
        cluster_size[(size_t)c * KVB_J + j] * KVB_DECAY + cnt * (1.f - KVB_DECAY);

    const float* __restrict__ ka = keys_avg + ((size_t)c * KVB_J + j) * KVB_DK;
    float* __restrict__ oka = out_ka + ((size_t)c * KVB_J + j) * KVB_DK;
#pragma unroll
    for (int k = 0; k < KVB_DK; ++k)
        oka[k] = ka[k] * KVB_DECAY + s[k] * (1.f - KVB_DECAY);
}

// ---------------------------------------------------------------------------
// Kernel 3: per-channel smoothing + new_keys = new_keys_avg / cs_smoothed.
// Grid: C. Block: 256.
// ---------------------------------------------------------------------------
__global__ __launch_bounds__(256) void kvb_finalize(
    const float* __restrict__ out_cs,
    const float* __restrict__ out_ka,
    float* __restrict__ out_keys)      // (C, J, dk)
{
    __shared__ float red[256];
    const int c = blockIdx.x;

    float local = 0.f;
    for (int j = threadIdx.x; j < KVB_J; j += 256)
        local += out_cs[(size_t)c * KVB_J + j];
    red[threadIdx.x] = local;
    __syncthreads();
    for (int step = 128; step > 0; step >>= 1) {
        if (threadIdx.x < step) red[threadIdx.x] += red[threadIdx.x + step];
        __syncthreads();
    }
    const float tot = red[0];

    for (int j = threadIdx.x; j < KVB_J; j += 256) {
        const float ncs = out_cs[(size_t)c * KVB_J + j];
        const float css = (ncs + KVB_EPS) / (tot + KVB_J * KVB_EPS) * tot;
        const float inv = 1.f / css;
        const float* __restrict__ ka = out_ka + ((size_t)c * KVB_J + j) * KVB_DK;
        float* __restrict__ ok = out_keys + ((size_t)c * KVB_J + j) * KVB_DK;
#pragma unroll
        for (int k = 0; k < KVB_DK; ++k) ok[k] = ka[k] * inv;
    }
}

// ---------------------------------------------------------------------------
extern "C" void kernel_launch(void* const* d_in, const int* in_sizes, int n_in,
                              void* d_out, int out_size, void* d_ws, size_t ws_size,
                              hipStream_t stream)
{
    (void)in_sizes; (void)n_in; (void)out_size; (void)ws_size;

    const float* x            = (const float*)d_in[0];
    const float* keys         = (const float*)d_in[1];
    const float* values       = (const float*)d_in[2];
    const float* keys_avg     = (const float*)d_in[3];
    const float* cluster_size = (const float*)d_in[4];

    float* out = (float*)d_out;
    // Output layout (flat, return order):
    const size_t qv_elems   = (size_t)KVB_B * KVB_C * 1 * KVB_N * KVB_DV; // 2,097,152
    const size_t keys_elems = (size_t)KVB_C * KVB_J * KVB_DK;             // 1,048,576
    const size_t cs_elems   = (size_t)KVB_C * KVB_J;                      //    65,536
    float* qv_out   = out;
    float* out_keys = out + qv_elems;
    float* out_cs   = out + qv_elems + keys_elems;
    float* out_ka   = out + qv_elems + keys_elems + cs_elems;

    int* ind_ws = (int*)d_ws;   // (C, I) = 512 KB

    kvb_assign<<<dim3(KVB_C, KVB_I / 128), 256, 0, stream>>>(
        x, keys, values, qv_out, ind_ws);
    kvb_segsum<<<dim3(KVB_C, KVB_J / 256), 256, 0, stream>>>(
        x, keys_avg, cluster_size, ind_ws, out_cs, out_ka);
    kvb_finalize<<<KVB_C, 256, 0, stream>>>(out_cs, out_ka, out_keys);
}